// Curvature_26637387169824
// MI455X (gfx1250) — compile-verified
//
#include <hip/hip_runtime.h>
#include <hip/hip_bf16.h>
#include <stdint.h>

// MI455X (gfx1250) implementation.
//
// Roofline: ~540 MB of unavoidable HBM traffic vs ~1.1 GFLOP of stencil math
// -> memory bound (~23 us at 23.3 TB/s). So the kernels are organized around
// CDNA5 data movement: ASYNCcnt-tracked global<->LDS B128 transfers
// (global_load_async_to_lds_b128 / global_store_async_from_lds_b128) for the
// streaming stages, and the matrix pipe (V_WMMA_F32_16X16X4_F32) for the
// cross-lane score reduction since it is otherwise idle.

typedef __attribute__((ext_vector_type(2))) float v2f;
typedef __attribute__((ext_vector_type(8))) float v8f;

#define B_   32
#define C_   512
#define HW_  64
#define K_   256          // int(0.5 * C)
#define PLANE_ (HW_ * HW_)   // 4096 floats = 16 KB

// ---------------------------------------------------------------------------
// Kernel 1: per-(b,c) curvature score.
// One 256-thread block per plane. Plane staged to LDS via async B128 loads.
// response = 5/16*(N+S+E+W) - 1/16*(NE+NW+SE+SW) - C ; score = sum |response|
// over the 62x62 VALID interior. Wave partials reduced with WMMA f32 16x16x4.
// ---------------------------------------------------------------------------
__global__ void curvature_score_kernel(const float* __restrict__ x,
                                       float* __restrict__ p) {
    __shared__ float sh[PLANE_];
    __shared__ float wred[8];

    const int plane = blockIdx.x;            // b*C + c
    const int t     = threadIdx.x;           // 0..255

    // --- async stage 16 KB plane into LDS: 4 x B128 per thread -------------
    const uint64_t src  = (uint64_t)(uintptr_t)(x + (size_t)plane * PLANE_);
    const unsigned goff = (unsigned)t * 64u;                   // bytes
    const unsigned lbase = (unsigned)(uintptr_t)(&sh[0]) + goff;
#pragma unroll
    for (int j = 0; j < 4; ++j) {
        unsigned la = lbase + (unsigned)j * 16u;
        unsigned ga = goff  + (unsigned)j * 16u;
        asm volatile("global_load_async_to_lds_b128 %0, %1, %2"
                     :: "v"(la), "v"(ga), "s"(src) : "memory");
    }
    asm volatile("s_wait_asynccnt 0x0" ::: "memory");
    __syncthreads();   // every wave has drained its own ASYNCcnt -> LDS ready

    // --- stencil: 64 cols x 4 row-stripes per thread -----------------------
    const int tx = t & 63;     // column 0..63
    const int ty = t >> 6;     // row phase 0..3
    float acc = 0.0f;
    if (tx >= 1 && tx <= 62) {
#pragma unroll 4
        for (int j = 1 + ty; j <= 62; j += 4) {
            const float* r0 = &sh[(j - 1) * HW_ + tx];
            const float* r1 = &sh[(j    ) * HW_ + tx];
            const float* r2 = &sh[(j + 1) * HW_ + tx];
            float nw = r0[-1], nn = r0[0], ne = r0[1];
            float ww = r1[-1], cc = r1[0], ee = r1[1];
            float sw = r2[-1], ss = r2[0], se = r2[1];
            float edge = (nn + ss) + (ee + ww);
            float corn = (nw + ne) + (sw + se);
            float r = fmaf(0.3125f, edge, fmaf(-0.0625f, corn, -cc));
            acc += fabsf(r);
        }
    }

    // --- wave(32) reduction on the matrix pipe -----------------------------
    // A(16x4): lane L<16 holds {m=L,k=0..1}, lane L+16 holds {m=L,k=2..3}.
    // a = {acc, 0} => A[m,0]=acc(lane m), A[m,2]=acc(lane m+16).
    // B = ones => D[m,n] = acc(lane m) + acc(lane m+16).
    // Sum of all 32 lane partials = sum_m D[m,n] = (8 VGPR adds) + shfl_xor(16).
    v2f a;  a.x = acc;  a.y = 0.0f;
    v2f ones; ones.x = 1.0f; ones.y = 1.0f;
    v8f cz = {};
    v8f d = __builtin_amdgcn_wmma_f32_16x16x4_f32(
        /*neg_a=*/false, a, /*neg_b=*/false, ones,
        /*c_mod=*/(short)0, cz, /*reuse_a=*/false, /*reuse_b=*/false);
    float s8 = ((d[0] + d[1]) + (d[2] + d[3])) + ((d[4] + d[5]) + (d[6] + d[7]));
    float tot = s8 + __shfl_xor(s8, 16, 32);

    const int wave = t >> 5;
    if ((t & 31) == 0) wred[wave] = tot;
    __syncthreads();
    if (t == 0) {
        float r = 0.0f;
#pragma unroll
        for (int i = 0; i < 8; ++i) r += wred[i];
        p[plane] = r;
    }
}

// ---------------------------------------------------------------------------
// Kernel 2: per-batch top-k by rank counting (matches lax.top_k ordering:
// value desc, stable index-asc ties). One block per batch row; 512^2 compares
// per row is negligible.
// ---------------------------------------------------------------------------
__global__ void topk_kernel(const float* __restrict__ p, int* __restrict__ idx) {
    __shared__ float pv[C_];
    const int b = blockIdx.x;
    const int t = threadIdx.x;   // 0..255

    pv[t]        = p[b * C_ + t];
    pv[t + 256]  = p[b * C_ + t + 256];
    __syncthreads();

#pragma unroll
    for (int ii = 0; ii < 2; ++ii) {
        const int i = t + ii * 256;
        const float mp = pv[i];
        int rank = 0;
        for (int j = 0; j < C_; ++j) {
            float pj = pv[j];
            rank += ((pj > mp) || (pj == mp && j < i)) ? 1 : 0;
        }
        if (rank < K_) idx[b * K_ + rank] = i;
    }
}

// ---------------------------------------------------------------------------
// Kernel 3: channel gather. One block per output plane; stream 16 KB through
// LDS on the async path (load_async_to_lds then store_async_from_lds). Each
// thread round-trips its own 64 bytes, so a per-wave s_wait_asynccnt is the
// only synchronization required.
// ---------------------------------------------------------------------------
__global__ void gather_kernel(const float* __restrict__ x,
                              const int* __restrict__ idx,
                              float* __restrict__ out) {
    __shared__ float sh[PLANE_];
    const int o = blockIdx.x;          // b*K + r
    const int b = o >> 8;              // K = 256
    const int c = idx[o];
    const int t = threadIdx.x;

    const uint64_t src = (uint64_t)(uintptr_t)(x + ((size_t)b * C_ + (size_t)c) * PLANE_);
    const uint64_t dst = (uint64_t)(uintptr_t)(out + (size_t)o * PLANE_);
    const unsigned goff  = (unsigned)t * 64u;
    const unsigned lbase = (unsigned)(uintptr_t)(&sh[0]) + goff;

#pragma unroll
    for (int j = 0; j < 4; ++j) {
        unsigned la = lbase + (unsigned)j * 16u;
        unsigned ga = goff  + (unsigned)j * 16u;
        asm volatile("global_load_async_to_lds_b128 %0, %1, %2"
                     :: "v"(la), "v"(ga), "s"(src) : "memory");
    }
    asm volatile("s_wait_asynccnt 0x0" ::: "memory");
#pragma unroll
    for (int j = 0; j < 4; ++j) {
        unsigned la = lbase + (unsigned)j * 16u;
        unsigned ga = goff  + (unsigned)j * 16u;
        asm volatile("global_store_async_from_lds_b128 %0, %1, %2"
                     :: "v"(ga), "v"(la), "s"(dst) : "memory");
    }
    asm volatile("s_wait_asynccnt 0x0" ::: "memory");
}

// ---------------------------------------------------------------------------
extern "C" void kernel_launch(void* const* d_in, const int* in_sizes, int n_in,
                              void* d_out, int out_size, void* d_ws, size_t ws_size,
                              hipStream_t stream) {
    (void)in_sizes; (void)n_in; (void)out_size; (void)ws_size;
    const float* x = (const float*)d_in[0];
    // d_in[1] is the fixed stencil; its values are baked into the score kernel.
    float* p   = (float*)d_ws;                                      // 32*512 f32 (64 KB)
    int*   idx = (int*)((char*)d_ws + (size_t)B_ * C_ * sizeof(float)); // 32*256 i32 (32 KB)
    float* out = (float*)d_out;

    curvature_score_kernel<<<B_ * C_, 256, 0, stream>>>(x, p);
    topk_kernel<<<B_, 256, 0, stream>>>(p, idx);
    gather_kernel<<<B_ * K_, 256, 0, stream>>>(x, idx, out);
}